// MambaBlock_32830730011528
// MI455X (gfx1250) — compile-verified
//
#include <hip/hip_runtime.h>
#include <hip/hip_bf16.h>
#include <math.h>

// ---- problem constants (match reference) ----
#define B_SZ     2
#define L_SZ     1024
#define DMODEL   1024
#define DSTATE   16
#define DCONV    4
#define DINNER   2048
#define DTRANK   64
#define BL       (B_SZ * L_SZ)          // 2048 token rows
#define XZ_COLS  (2 * DINNER)           // 4096
#define XDBL_COLS (DTRANK + 2 * DSTATE) // 96
#define KB       64                     // K-panel staged in LDS per iteration

typedef float v2f __attribute__((ext_vector_type(2)));
typedef float v8f __attribute__((ext_vector_type(8)));
typedef unsigned int u32x4 __attribute__((ext_vector_type(4)));
typedef int i32x4 __attribute__((ext_vector_type(4)));
typedef int i32x8 __attribute__((ext_vector_type(8)));

#if defined(__gfx1250__) && __has_builtin(__builtin_amdgcn_tensor_load_to_lds) && __has_builtin(__builtin_amdgcn_s_wait_tensorcnt)
#define USE_TDM 1
#else
#define USE_TDM 0
#endif

// f32 16x16x4 WMMA fragment layout (ISA 7.12.2):
//   A (16x4): lane<16 -> M=lane, K={0,1}; lane>=16 -> K={2,3}
//   B (4x16): N=lane&15, (lane>>4) selects K pair
//   C (16x16): vgpr r, lane -> M = r + (lane>>4)*8, N = lane&15
__device__ __forceinline__ v8f wmma4(v2f a, v2f b, v8f c) {
    return __builtin_amdgcn_wmma_f32_16x16x4_f32(false, a, false, b, (short)0, c, false, false);
}

__device__ __forceinline__ float softplus_f(float v) {
    return (v > 20.f) ? v : log1pf(__expf(v));
}

#if USE_TDM
// Issue one TDM transfer: B[kb:kb+KB, nBase:nBase+64] (row-major, 4B elems)
// into an LDS panel.  2D D#: tile 64 x KB, dim0 stride = ldb.
__device__ __forceinline__ void tdm_load_panel(const float* src, int ldb, int K,
                                               float* dstLds) {
    unsigned long long ga = (unsigned long long)(uintptr_t)src;
    unsigned int ldsOff = (unsigned int)(uintptr_t)dstLds;
    u32x4 g0;                                    // count=1 | lds | global | type=2
    g0.x = 1u;
    g0.y = ldsOff;
    g0.z = (unsigned int)ga;
    g0.w = (unsigned int)((ga >> 32) & 0x01FFFFFFull) | (2u << 30);
    const unsigned int td0 = (unsigned int)ldb;  // tensor_dim0
    const unsigned int td1 = (unsigned int)K;    // tensor_dim1
    const unsigned long long s0 = (unsigned long long)(unsigned int)ldb;
    i32x8 g1;
    g1[0] = (int)(2u << 16);                                // data_size=2 (4B)
    g1[1] = (int)((td0 & 0xFFFFu) << 16);                   // dim0[15:0]
    g1[2] = (int)((td0 >> 16) | ((td1 & 0xFFFFu) << 16));   // dim0[31:16]|dim1[15:0]
    g1[3] = (int)((td1 >> 16) | (64u << 16));               // dim1[31:16]|tile0=64
    g1[4] = (int)KB;                                        // tile1=KB, tile2=0
    g1[5] = (int)(s0 & 0xFFFFFFFFull);                      // stride0 lo
    g1[6] = (int)((s0 >> 32) & 0xFFFFull);                  // stride0 hi
    g1[7] = 0;
    i32x4 z4 = {0, 0, 0, 0};
#if __clang_major__ >= 23
    i32x8 z8 = {0, 0, 0, 0, 0, 0, 0, 0};
    __builtin_amdgcn_tensor_load_to_lds(g0, g1, z4, z4, z8, 0);
#else
    __builtin_amdgcn_tensor_load_to_lds(g0, g1, z4, z4, 0);
#endif
}
#endif

// ---------------------------------------------------------------------------
// LDS/TDM double-buffered fp32 WMMA GEMM.  Requires: M % 128 == 0,
// N % 64 == 0, K % KB == 0.  Grid: (M/128, N/64).  8 waves/block; wave w owns
// M-tile blockIdx.x*8+w; all waves share a 64x64 B panel.  Panels ping-pong:
// the TDM fill of panel i+1 overlaps the 64 WMMAs consuming panel i
// (s_wait_tensorcnt(1) = wait only for the oldest in-flight transfer).
// MODE 0: plain store. MODE 1: softplus(v + bias[col]).
// ---------------------------------------------------------------------------
template<int MODE>
__global__ __launch_bounds__(256) void gemm_wmma_lds(
    const float* __restrict__ A, int lda,
    const float* __restrict__ Bm, int ldb,
    float* __restrict__ C, int ldc,
    int K, const float* __restrict__ bias)
{
    __shared__ float bpanel[2][KB * 64];   // 2 x 16 KB

    const int lane = threadIdx.x & 31;
    const int wave = threadIdx.x >> 5;
    const int tM    = blockIdx.x * 8 + wave;
    const int nBase = blockIdx.y * 64;

    const int aRow = (tM << 4) + (lane & 15);
    const int kSub = (lane >> 4) << 1;          // 0 or 2
    const int nCol = lane & 15;

    v8f acc[4] = {};

    const int nIter = K / KB;

#if USE_TDM
    if (wave == 0)
        tdm_load_panel(Bm + nBase, ldb, K, &bpanel[0][0]);   // prologue: panel 0
#endif

    for (int it = 0; it < nIter; ++it) {
        const int kb = it * KB;
        const float* cur = &bpanel[it & 1][0];

#if USE_TDM
        if (wave == 0) {
            if (it + 1 < nIter) {
                tdm_load_panel(Bm + (size_t)(kb + KB) * ldb + nBase, ldb, K,
                               &bpanel[(it + 1) & 1][0]);
                __builtin_amdgcn_s_wait_tensorcnt(1);  // current panel done; next in flight
            } else {
                __builtin_amdgcn_s_wait_tensorcnt(0);
            }
        }
#else
        {
            float* dst = &bpanel[it & 1][0];
            for (int i = threadIdx.x; i < KB * 64; i += 256) {
                const int r = i >> 6, c = i & 63;
                dst[i] = Bm[(size_t)(kb + r) * ldb + nBase + c];
            }
        }
#endif
        __syncthreads();                       // panel `cur` visible to all waves

        const float* aPtr = A + (size_t)aRow * lda + kb + kSub;
        __builtin_prefetch(aPtr + KB, 0, 1);   // speculative: next A panel

        #pragma unroll 4
        for (int kk = 0; kk < KB; kk += 4) {
            v2f a = *(const v2f*)(aPtr + kk);
            const float* b0 = &cur[(kk + kSub) * 64 + nCol];
            const float* b1 = b0 + 64;
            #pragma unroll
            for (int j = 0; j < 4; ++j) {
                v2f b; b.x = b0[j * 16]; b.y = b1[j * 16];
                acc[j] = wmma4(a, b, acc[j]);
            }
        }
        __syncthreads();                       // all waves done with `cur` before reuse
    }

    const int rowBase = (tM << 4) + ((lane >> 4) << 3);
    #pragma unroll
    for (int j = 0; j < 4; ++j) {
        const int col = nBase + nCol + (j << 4);
        const float bv = (MODE == 1) ? bias[col] : 0.f;
        #pragma unroll
        for (int r = 0; r < 8; ++r) {
            float v = acc[j][r];
            if (MODE == 1) v = softplus_f(v + bv);
            C[(size_t)(rowBase + r) * ldc + col] = v;
        }
    }
}

// ---------------------------------------------------------------------------
// Direct fp32 WMMA GEMM for skinny N (N % (NT*16) == 0), e.g. N=96 with NT=2.
// Compile-time NT -> no exec-mask games around the WMMAs.
// ---------------------------------------------------------------------------
template<int NT, int MODE>
__global__ __launch_bounds__(256) void gemm_wmma_direct(
    const float* __restrict__ A, int lda,
    const float* __restrict__ Bm, int ldb,
    float* __restrict__ C, int ldc,
    int M, int N, int K, const float* __restrict__ bias)
{
    const int lane = threadIdx.x & 31;
    const int wave = threadIdx.x >> 5;
    const int tilesM  = M >> 4;
    const int nGroups = (N >> 4) / NT;

    const int wid = blockIdx.x * 8 + wave;
    if (wid >= tilesM * nGroups) return;        // wave-uniform

    const int gN = wid % nGroups;
    const int tM = wid / nGroups;

    const int aRow    = (tM << 4) + (lane & 15);
    const int kSub    = (lane >> 4) << 1;
    const int colBase = gN * (NT * 16) + (lane & 15);

    v8f acc[NT] = {};

    for (int k0 = 0; k0 < K; k0 += 4) {
        v2f a = *(const v2f*)(A + (size_t)aRow * lda + k0 + kSub);
        __builtin_prefetch(A + (size_t)aRow * lda + k0 + 16 + kSub, 0, 1);
        const float* b0 = Bm + (size_t)(k0 + kSub) * ldb + colBase;
        const float* b1 = b0 + ldb;
        #pragma unroll
        for (int j = 0; j < NT; ++j) {
            v2f b; b.x = b0[j * 16]; b.y = b1[j * 16];
            acc[j] = wmma4(a, b, acc[j]);
        }
    }

    const int rowBase = (tM << 4) + ((lane >> 4) << 3);
    #pragma unroll
    for (int j = 0; j < NT; ++j) {
        const int col = colBase + (j << 4);
        const float bv = (MODE == 1) ? bias[col] : 0.f;
        #pragma unroll
        for (int r = 0; r < 8; ++r) {
            float v = acc[j][r];
            if (MODE == 1) v = softplus_f(v + bv);
            C[(size_t)(rowBase + r) * ldc + col] = v;
        }
    }
}

// ---------------------------------------------------------------------------
// Depthwise causal conv (width 4) + bias + SiLU over the u-half of xz.
// ---------------------------------------------------------------------------
__global__ __launch_bounds__(256) void conv_silu(
    const float* __restrict__ xz, const float* __restrict__ conv_w,
    const float* __restrict__ conv_b, float* __restrict__ ucv)
{
    int i = blockIdx.x * 256 + threadIdx.x;
    if (i >= BL * DINNER) return;
    const int d   = i & (DINNER - 1);
    const int idx = i >> 11;            // token row (b*L + l)
    const int l   = idx & (L_SZ - 1);
    const int b   = idx >> 10;

    float acc = conv_b[d];
    #pragma unroll
    for (int k = 0; k < DCONV; ++k) {
        const int ls = l - (DCONV - 1) + k;
        if (ls >= 0)
            acc += xz[(size_t)(b * L_SZ + ls) * XZ_COLS + d] * conv_w[d * DCONV + k];
    }
    ucv[(size_t)idx * DINNER + d] = acc * (1.f / (1.f + __expf(-acc)));
}

// ---------------------------------------------------------------------------
// Selective scan, one thread per (b,d) channel; 16 states in registers.
// Fuses y = (scan + u*D_skip) * silu(z) epilogue.
// ---------------------------------------------------------------------------
__global__ __launch_bounds__(256) void scan_fused(
    const float* __restrict__ delta, const float* __restrict__ ucv,
    const float* __restrict__ xdbl,  const float* __restrict__ xz,
    const float* __restrict__ A_log, const float* __restrict__ D_skip,
    float* __restrict__ ygate)
{
    int t = blockIdx.x * 256 + threadIdx.x;
    if (t >= B_SZ * DINNER) return;
    const int b = t / DINNER;
    const int d = t % DINNER;

    float A[DSTATE], h[DSTATE];
    #pragma unroll
    for (int n = 0; n < DSTATE; ++n) { A[n] = -__expf(A_log[d * DSTATE + n]); h[n] = 0.f; }
    const float dsk = D_skip[d];

    for (int l = 0; l < L_SZ; ++l) {
        const int idx = b * L_SZ + l;
        const float dlt = delta[(size_t)idx * DINNER + d];
        const float uu  = ucv[(size_t)idx * DINNER + d];
        const float du  = dlt * uu;
        const float* Brow = xdbl + (size_t)idx * XDBL_COLS + DTRANK;
        const float* Crow = Brow + DSTATE;
        float y = 0.f;
        #pragma unroll
        for (int n = 0; n < DSTATE; ++n) {
            const float dA = __expf(dlt * A[n]);
            h[n] = dA * h[n] + du * Brow[n];
            y += h[n] * Crow[n];
        }
        const float zv = xz[(size_t)idx * XZ_COLS + DINNER + d];
        const float sz = zv * (1.f / (1.f + __expf(-zv)));
        ygate[(size_t)idx * DINNER + d] = (y + uu * dsk) * sz;
    }
}

// ---------------------------------------------------------------------------
extern "C" void kernel_launch(void* const* d_in, const int* in_sizes, int n_in,
                              void* d_out, int out_size, void* d_ws, size_t ws_size,
                              hipStream_t stream) {
    (void)in_sizes; (void)n_in; (void)out_size; (void)ws_size;
    const float* x      = (const float*)d_in[0];
    const float* W_in   = (const float*)d_in[1];
    const float* conv_w = (const float*)d_in[2];
    const float* conv_b = (const float*)d_in[3];
    const float* W_x    = (const float*)d_in[4];
    const float* W_dt   = (const float*)d_in[5];
    const float* b_dt   = (const float*)d_in[6];
    const float* A_log  = (const float*)d_in[7];
    const float* D_skip = (const float*)d_in[8];
    const float* W_out  = (const float*)d_in[9];
    float* out = (float*)d_out;

    // workspace carve-out (floats): ~85 MB
    float* ws    = (float*)d_ws;
    float* xz    = ws;                               // BL*4096
    float* ucv   = xz    + (size_t)BL * XZ_COLS;     // BL*2048
    float* xdbl  = ucv   + (size_t)BL * DINNER;      // BL*96
    float* delta = xdbl  + (size_t)BL * XDBL_COLS;   // BL*2048
    float* ygate = delta + (size_t)BL * DINNER;      // BL*2048

    const dim3 blk(256);

    // 1) xz = x @ W_in                [2048,1024]x[1024,4096]  (TDM/LDS kernel)
    gemm_wmma_lds<0><<<dim3(BL / 128, XZ_COLS / 64), blk, 0, stream>>>(
        x, DMODEL, W_in, XZ_COLS, xz, XZ_COLS, DMODEL, nullptr);

    // 2) u = silu(causal_conv(xz[:, :2048]) + conv_b)
    conv_silu<<<(BL * DINNER) / 256, blk, 0, stream>>>(xz, conv_w, conv_b, ucv);

    // 3) x_dbl = u @ W_x              [2048,2048]x[2048,96]    (direct, NT=2)
    {
        const int waves  = (BL >> 4) * ((XDBL_COLS >> 4) / 2);
        gemm_wmma_direct<2, 0><<<(waves + 7) / 8, blk, 0, stream>>>(
            ucv, DINNER, W_x, XDBL_COLS, xdbl, XDBL_COLS, BL, XDBL_COLS, DINNER, nullptr);
    }

    // 4) delta = softplus(x_dbl[:, :64] @ W_dt + b_dt)  [2048,64]x[64,2048]
    gemm_wmma_lds<1><<<dim3(BL / 128, DINNER / 64), blk, 0, stream>>>(
        xdbl, XDBL_COLS, W_dt, DINNER, delta, DINNER, DTRANK, b_dt);

    // 5) selective scan + skip + silu(z) gating
    scan_fused<<<(B_SZ * DINNER) / 256, blk, 0, stream>>>(
        delta, ucv, xdbl, xz, A_log, D_skip, ygate);

    // 6) out = ygate @ W_out          [2048,2048]x[2048,1024]  (TDM/LDS kernel)
    gemm_wmma_lds<0><<<dim3(BL / 128, DMODEL / 64), blk, 0, stream>>>(
        ygate, DINNER, W_out, DMODEL, out, DMODEL, DINNER, nullptr);
}